// AutoCon_2241972929072
// MI455X (gfx1250) — compile-verified
//
#include <hip/hip_runtime.h>
#include <hip/hip_bf16.h>
#include <math.h>

// Problem constants (from reference setup_inputs)
#define BB   64
#define TT   1536
#define DD   512
#define TS   512            // T // 3
#define ACFN 5000
#define INV_T (1.0f / 0.07f)
#define F_NEG_INF (-__builtin_inff())

typedef __attribute__((ext_vector_type(2))) float v2f;
typedef __attribute__((ext_vector_type(8))) float v8f;

// Workspace layout (floats)
//   logits : [64][512][512]  -> 16,777,216 floats
//   pooled : [64][512]       ->     32,768 floats
//   colmax : [64][512]       ->     32,768 floats
#define WS_LOGITS 0
#define WS_POOLED (64 * 512 * 512)
#define WS_COLMAX (WS_POOLED + 64 * 512)

// ---------------------------------------------------------------------------
// 1) Max-pool over T:  pooled[b,d] = max_t features[b,t,d]
// ---------------------------------------------------------------------------
__global__ __launch_bounds__(256)
void autocon_pool_max(const float* __restrict__ feat, float* __restrict__ pooled) {
  int t = blockIdx.x * 256 + threadIdx.x;      // 0 .. 64*512-1
  int b = t >> 9;
  int d = t & 511;
  const float* p = feat + (size_t)b * TT * DD + d;
  float m = F_NEG_INF;
  for (int tt = 0; tt < TT; ++tt) m = fmaxf(m, p[(size_t)tt * DD]);
  pooled[t] = m;
}

// ---------------------------------------------------------------------------
// 2) Batched GEMM via V_WMMA_F32_16X16X4_F32:
//    logits[b,i,j] = (1/T) * sum_d sel[b,i,d] * sel[b,j,d]
//    One wave computes a 64x64 tile: 16 v8f accumulators, 16 WMMA per K-step
//    against 8 float2 gathered loads (2:1 wmma:vmem in steady state).
//    blockIdx.y = batch, so the feature base pointer is block-uniform and
//    gathers use 32-bit per-lane offsets (SGPR base + VGPR offset form).
// ---------------------------------------------------------------------------
__global__ __launch_bounds__(128)
void autocon_gemm_logits(const float* __restrict__ feat,
                         const int* __restrict__ idxs,
                         float* __restrict__ logits) {
  const int wave = threadIdx.x >> 5;
  const int lane = threadIdx.x & 31;
  const int half = lane >> 4;     // 0: lanes 0-15, 1: lanes 16-31
  const int lm   = lane & 15;

  const int b = blockIdx.y;                    // batch (block-uniform)
  int t  = blockIdx.x * 4 + wave;              // tile id 0..63 (8x8 grid)
  int i0 = (t >> 3) << 6;                      // row-tile origin (x64)
  int j0 = (t & 7) << 6;                       // col-tile origin (x64)

  const int*   idxp = idxs + b * TS;
  const float* base = feat + (size_t)b * TT * DD;   // uniform per block

  // A-fragment f32 16x4 layout: lanes 0-15 hold M=lm, K={k,k+1};
  // lanes 16-31 hold M=lm, K={k+2,k+3} -> float2 load at column 2*half.
  int offA[4], offB[4];
#pragma unroll
  for (int f = 0; f < 4; ++f) {
    offA[f] = idxp[i0 + 16 * f + lm] * DD + 2 * half;
    offB[f] = idxp[j0 + 16 * f + lm] * DD + 2 * half;
  }

  v8f acc[4][4] = {};

  for (int k = 0; k < DD; k += 4) {
    v2f av[4], bv[4];
#pragma unroll
    for (int f = 0; f < 4; ++f) {
      av[f] = *(const v2f*)(base + offA[f] + k);
      bv[f] = *(const v2f*)(base + offB[f] + k);
    }
#pragma unroll
    for (int mi = 0; mi < 4; ++mi)
#pragma unroll
      for (int ni = 0; ni < 4; ++ni)
        acc[mi][ni] = __builtin_amdgcn_wmma_f32_16x16x4_f32(
            false, av[mi], false, bv[ni], (short)0, acc[mi][ni], false, false);
  }

  // C/D layout: VGPR r -> row M = r + 8*half, col N = lm.
  float* lb = logits + (size_t)b * TS * TS;
#pragma unroll
  for (int mi = 0; mi < 4; ++mi) {
#pragma unroll
    for (int ni = 0; ni < 4; ++ni) {
#pragma unroll
      for (int r = 0; r < 8; ++r) {
        int row = i0 + 16 * mi + r + 8 * half;
        int col = j0 + 16 * ni + lm;
        lb[(size_t)row * TS + col] = acc[mi][ni][r] * INV_T;
      }
    }
  }
}

// ---------------------------------------------------------------------------
// 3) Column max: colmax[b,j] = max_i logits[b,i,j]  (ref: max over axis=1)
// ---------------------------------------------------------------------------
__global__ __launch_bounds__(256)
void autocon_col_max(const float* __restrict__ logits, float* __restrict__ colmax) {
  int t = blockIdx.x * 256 + threadIdx.x;      // b*512 + j
  int b = t >> 9;
  int j = t & 511;
  const float* p = logits + (size_t)b * TS * TS + j;
  float m = F_NEG_INF;
  for (int i = 0; i < TS; ++i) m = fmaxf(m, p[(size_t)i * TS]);
  colmax[t] = m;
}

// ---------------------------------------------------------------------------
// 4) Local loss row pass: one wave per (b, i) row.
// ---------------------------------------------------------------------------
__global__ __launch_bounds__(256)
void autocon_local_row(const float* __restrict__ logits,
                       const float* __restrict__ colmax,
                       const int* __restrict__ idxs,
                       const float* __restrict__ acf,
                       float* __restrict__ out) {
  int wave = threadIdx.x >> 5;
  int lane = threadIdx.x & 31;
  int r = blockIdx.x * 8 + wave;               // 0 .. 64*512-1
  int b = r >> 9;
  int i = r & 511;

  const float* lrow = logits + (size_t)r * TS;
  const float* cm   = colmax + b * TS;
  const int*   ip   = idxs + b * TS;
  int idx_i = ip[i];

  float lv[16], dv[16];
  float dmax = F_NEG_INF, S = 0.0f;
#pragma unroll
  for (int it = 0; it < 16; ++it) {
    int j = lane + it * 32;
    float l = lrow[j] - cm[j];                 // logits shifted by column max
    int dd = ip[j] - idx_i; if (dd < 0) dd = -dd;
    float d = acf[dd];
    lv[it] = l; dv[it] = d;
    bool self = (d == 1.0f);
    float masked = self ? F_NEG_INF : d;
    dmax = fmaxf(dmax, masked);
    if (j != i) S += expf(l);                  // exp_logits * neg_mask
  }
#pragma unroll
  for (int o = 16; o; o >>= 1) {
    dmax = fmaxf(dmax, __shfl_xor(dmax, o, 32));
    S += __shfl_xor(S, o, 32);
  }
  float logS = logf(S);

  float Wsum = 0.0f, wsum = 0.0f, n = 0.0f;
#pragma unroll
  for (int it = 0; it < 16; ++it) {
    int j = lane + it * 32;
    float d = dv[it];
    bool self = (d == 1.0f);
    bool neg  = (j != i);
    float masked = self ? F_NEG_INF : d;
    float pos = ((masked == dmax) ? 1.0f : 0.0f) + ((neg && self) ? 1.0f : 0.0f);
    float w = d * pos;                         // distmap * pos_mask
    n += pos; wsum += w; Wsum += w * lv[it];
  }
#pragma unroll
  for (int o = 16; o; o >>= 1) {
    n    += __shfl_xor(n, o, 32);
    wsum += __shfl_xor(wsum, o, 32);
    Wsum += __shfl_xor(Wsum, o, 32);
  }
  if (lane == 0) out[r] = -(Wsum - logS * wsum) / n;   // -(T/BT)=-1 factor
}

// ---------------------------------------------------------------------------
// 5) Global loss: single block. logits64 = pooled @ pooled^T / T, then the
//    same mask/softmax logic with labels-based distmap.
// ---------------------------------------------------------------------------
__global__ __launch_bounds__(256)
void autocon_global_loss(const float* __restrict__ pooled,
                         const float* __restrict__ acf,
                         const int* __restrict__ labels,
                         float* __restrict__ out) {
  __shared__ float lg[64][64];
  __shared__ float cmax[64];
  int tid = threadIdx.x;

  for (int p = tid; p < 64 * 64; p += 256) {
    int i = p >> 6, j = p & 63;
    const float* pi = pooled + i * DD;
    const float* pj = pooled + j * DD;
    float s = 0.0f;
    for (int d = 0; d < DD; ++d) s += pi[d] * pj[d];
    lg[i][j] = s * INV_T;
  }
  __syncthreads();

  if (tid < 64) {
    float m = F_NEG_INF;
    for (int i = 0; i < 64; ++i) m = fmaxf(m, lg[i][tid]);
    cmax[tid] = m;
  }
  __syncthreads();

  if (tid < 64) {
    int i = tid;
    int li = labels[i];
    float dmax = F_NEG_INF, S = 0.0f;
    for (int j = 0; j < 64; ++j) {
      int dd = labels[j] - li; if (dd < 0) dd = -dd;
      float d = acf[dd];
      bool self = (d == 1.0f);
      float masked = self ? F_NEG_INF : d;
      dmax = fmaxf(dmax, masked);
      if (j != i) S += expf(lg[i][j] - cmax[j]);
    }
    float logS = logf(S);
    float Wsum = 0.0f, wsum = 0.0f, n = 0.0f;
    for (int j = 0; j < 64; ++j) {
      int dd = labels[j] - li; if (dd < 0) dd = -dd;
      float d = acf[dd];
      bool self = (d == 1.0f);
      bool neg  = (j != i);
      float masked = self ? F_NEG_INF : d;
      float pos = ((masked == dmax) ? 1.0f : 0.0f) + ((neg && self) ? 1.0f : 0.0f);
      float w = d * pos;
      n += pos; wsum += w; Wsum += w * (lg[i][j] - cmax[j]);
    }
    out[64 * 512 + i] = -(Wsum - logS * wsum) / n;
  }
}

// ---------------------------------------------------------------------------
extern "C" void kernel_launch(void* const* d_in, const int* in_sizes, int n_in,
                              void* d_out, int out_size, void* d_ws, size_t ws_size,
                              hipStream_t stream) {
  const float* features = (const float*)d_in[0];   // [64,1536,512] f32
  const float* acf      = (const float*)d_in[1];   // [5000] f32
  const int*   labels   = (const int*)d_in[2];     // [64] i32
  const int*   fidx     = (const int*)d_in[3];     // [64,512] i32

  float* ws     = (float*)d_ws;
  float* logits = ws + WS_LOGITS;
  float* pooled = ws + WS_POOLED;
  float* colmax = ws + WS_COLMAX;
  float* out    = (float*)d_out;                   // [64*512 local | 64 global]

  // 1) pooled[b,d] = max_t features[b,t,d]
  autocon_pool_max<<<(BB * DD) / 256, 256, 0, stream>>>(features, pooled);

  // 2) batched 512x512x512 fp32 WMMA GEMM -> logits (already /TEMPERATURE)
  //    grid: 16 blocks of 4 waves per batch (64 tiles of 64x64), y = batch
  autocon_gemm_logits<<<dim3(16, BB), 128, 0, stream>>>(features, fidx, logits);

  // 3) column max per (b, j)
  autocon_col_max<<<(BB * TS) / 256, 256, 0, stream>>>(logits, colmax);

  // 4) local loss rows (one wave per row)
  autocon_local_row<<<(BB * TS) / 8, 256, 0, stream>>>(logits, colmax, fidx, acf, out);

  // 5) global loss (single block)
  autocon_global_loss<<<1, 256, 0, stream>>>(pooled, acf, labels, out);
}